// GCN_87299505258672
// MI455X (gfx1250) — compile-verified
//
#include <hip/hip_runtime.h>

#define NNODES 40000
#define NEDGES 640000
#define NGRAPH 64
#define COUT   10
#define BN_INV_F 0.99999500003749973f   // 1/sqrt(1 + 1e-5)

typedef __attribute__((ext_vector_type(2))) float v2f;
typedef __attribute__((ext_vector_type(8))) float v8f;

// ---------------------------------------------------------------- utilities

__global__ void k_count(const int* __restrict__ batch, float* __restrict__ cnt) {
    int i = blockIdx.x * blockDim.x + threadIdx.x;
    if (i < NNODES) unsafeAtomicAdd(&cnt[batch[i]], 1.0f);
}

// float4 copy (n must be a multiple of 4)
__global__ void k_copy4(float4* __restrict__ dst, const float4* __restrict__ src, int n4) {
    int i = blockIdx.x * blockDim.x + threadIdx.x;
    if (i < n4) dst[i] = src[i];
}

// agg[dst[e]][c] += x[src[e]][c]   (agg pre-initialized to x => h = x + sum)
template <int C>
__global__ void k_scatter_add(const float* __restrict__ x,
                              const int* __restrict__ srcs,
                              const int* __restrict__ dsts,
                              float* __restrict__ agg) {
    long long t = (long long)blockIdx.x * blockDim.x + threadIdx.x;
    if (t >= (long long)NEDGES * C) return;
    int e = (int)(t / C);
    int c = (int)(t - (long long)e * C);
    int s = srcs[e], d = dsts[e];
    unsafeAtomicAdd(&agg[(long long)d * C + c], x[(long long)s * C + c]);
}

// pooled[batch[i]][c] += h[i][c]   (256-wide h)
__global__ void k_pool(const float* __restrict__ h,
                       const int* __restrict__ batch,
                       float* __restrict__ pooled) {
    long long t = (long long)blockIdx.x * blockDim.x + threadIdx.x;
    if (t >= (long long)NNODES * 256) return;
    int i = (int)(t >> 8);
    int c = (int)(t & 255);
    unsafeAtomicAdd(&pooled[batch[i] * 256 + c], h[t]);
}

// ------------------------------------------------- fused GEMM + BN + ReLU
// out[N x 256] = relu( gamma*BN_INV*(A[N x K] @ W[K x 256] + bias) + beta )
// Block = 64 rows x 256 cols (grid = 40000/64 = 625 exact), 8 waves.
// Wave w owns col tiles {2w, 2w+1} and iterates 4 row tiles, so each B
// fragment (global) feeds 4 WMMAs -> 8 WMMAs per 4 global loads per k-step.
// A slab staged in LDS: 64 x 260 floats (66.5 KB), conflict-free stride.
template <int K>
__global__ __launch_bounds__(256)
void k_gemm_bn_relu(const float* __restrict__ A,
                    const float* __restrict__ W,      // [K][256] row-major
                    const float* __restrict__ bias,
                    const float* __restrict__ gamma,
                    const float* __restrict__ beta,
                    float* __restrict__ out) {
    __shared__ float As[64][260];

    const int row0 = blockIdx.x * 64;
    const int tid  = threadIdx.x;
    const int lane = tid & 31;
    const int wave = tid >> 5;

    // cooperative float4 load of the 64 x K activation slab
    constexpr int KV = K / 4;
    const float4* A4 = reinterpret_cast<const float4*>(A);
    for (int i = tid; i < 64 * KV; i += 256) {
        int m = i / KV, kv = i - m * KV;
        float4 v = A4[(long long)(row0 + m) * KV + kv];
        *reinterpret_cast<float4*>(&As[m][kv * 4]) = v;
    }
    __syncthreads();

    const int mlane = lane & 15;               // row (A) / col (B) within tile
    const int khalf = (lane >> 4) << 1;        // lanes 16-31 carry K+2,K+3
    const int n0    = (wave * 2) * 16 + mlane;
    const int n1    = n0 + 16;

    v8f acc[4][2];
    {
        v8f z = {};
#pragma unroll
        for (int rt = 0; rt < 4; ++rt) { acc[rt][0] = z; acc[rt][1] = z; }
    }

#pragma unroll 4
    for (int k0 = 0; k0 < K; k0 += 4) {
        const int krow = k0 + khalf;
        v2f b0, b1;
        b0.x = W[krow * 256 + n0];
        b0.y = W[(krow + 1) * 256 + n0];
        b1.x = W[krow * 256 + n1];
        b1.y = W[(krow + 1) * 256 + n1];
#pragma unroll
        for (int rt = 0; rt < 4; ++rt) {
            v2f a;
            a.x = As[rt * 16 + mlane][krow];
            a.y = As[rt * 16 + mlane][krow + 1];
            acc[rt][0] = __builtin_amdgcn_wmma_f32_16x16x4_f32(
                false, a, false, b0, (short)0, acc[rt][0], false, false);
            acc[rt][1] = __builtin_amdgcn_wmma_f32_16x16x4_f32(
                false, a, false, b1, (short)0, acc[rt][1], false, false);
        }
    }

    // epilogue: VGPR r -> M = r + 8*(lane>=16), N = col tile base + (lane&15)
    const int mbase = (lane >> 4) * 8;
    const float g0 = gamma[n0] * BN_INV_F, bb0 = beta[n0], bi0 = bias[n0];
    const float g1 = gamma[n1] * BN_INV_F, bb1 = beta[n1], bi1 = bias[n1];
#pragma unroll
    for (int rt = 0; rt < 4; ++rt) {
        const long long rowb = row0 + rt * 16 + mbase;
#pragma unroll
        for (int r = 0; r < 8; ++r) {
            float v0 = fmaf(g0, acc[rt][0][r] + bi0, bb0);
            float v1 = fmaf(g1, acc[rt][1][r] + bi1, bb1);
            out[(rowb + r) * 256 + n0] = v0 > 0.f ? v0 : 0.f;
            out[(rowb + r) * 256 + n1] = v1 > 0.f ? v1 : 0.f;
        }
    }
}

// ------------------------------------------------------------- readout head
// out[g][o] = lin_b[o] + sum_c (p0[g][c]/cnt)*W[c][o] + (p1[g][c]/cnt)*W[256+c][o]
__global__ void k_head(const float* __restrict__ p0, const float* __restrict__ p1,
                       const float* __restrict__ cnt,
                       const float* __restrict__ W,   // [512][10]
                       const float* __restrict__ b,
                       float* __restrict__ out) {
    int g = threadIdx.x;
    if (g >= NGRAPH) return;
    float inv = 1.0f / fmaxf(cnt[g], 1.0f);
    float acc[COUT];
#pragma unroll
    for (int o = 0; o < COUT; ++o) acc[o] = b[o];
    for (int c = 0; c < 256; ++c) {
        float v = p0[g * 256 + c] * inv;
#pragma unroll
        for (int o = 0; o < COUT; ++o) acc[o] = fmaf(v, W[c * COUT + o], acc[o]);
    }
    for (int c = 0; c < 256; ++c) {
        float v = p1[g * 256 + c] * inv;
#pragma unroll
        for (int o = 0; o < COUT; ++o) acc[o] = fmaf(v, W[(256 + c) * COUT + o], acc[o]);
    }
#pragma unroll
    for (int o = 0; o < COUT; ++o) out[g * COUT + o] = acc[o];
}

// --------------------------------------------------------------------------

extern "C" void kernel_launch(void* const* d_in, const int* in_sizes, int n_in,
                              void* d_out, int out_size, void* d_ws, size_t ws_size,
                              hipStream_t stream) {
    const float* x     = (const float*)d_in[0];
    const int*   ei    = (const int*)d_in[1];
    const int*   src   = ei;
    const int*   dst   = ei + NEDGES;
    const int*   batch = (const int*)d_in[2];

    const float* l0_W1 = (const float*)d_in[3];
    const float* l0_b1 = (const float*)d_in[4];
    const float* l0_g1 = (const float*)d_in[5];
    const float* l0_be1= (const float*)d_in[6];
    const float* l0_W2 = (const float*)d_in[7];
    const float* l0_b2 = (const float*)d_in[8];
    const float* l0_g2 = (const float*)d_in[9];
    const float* l0_be2= (const float*)d_in[10];
    const float* l1_W1 = (const float*)d_in[11];
    const float* l1_b1 = (const float*)d_in[12];
    const float* l1_g1 = (const float*)d_in[13];
    const float* l1_be1= (const float*)d_in[14];
    const float* l1_W2 = (const float*)d_in[15];
    const float* l1_b2 = (const float*)d_in[16];
    const float* l1_g2 = (const float*)d_in[17];
    const float* l1_be2= (const float*)d_in[18];
    const float* lin_W = (const float*)d_in[19];
    const float* lin_b = (const float*)d_in[20];

    float* ws   = (float*)d_ws;
    float* buf0 = ws;                                   // [N][256]
    float* buf1 = ws + (size_t)NNODES * 256;            // [N][256]
    float* buf2 = ws + (size_t)NNODES * 256 * 2;        // [N][256]
    float* p0   = ws + (size_t)NNODES * 256 * 3;        // [G][256]
    float* p1   = p0 + NGRAPH * 256;                    // [G][256]
    float* cnt  = p1 + NGRAPH * 256;                    // [G]

    hipMemsetAsync(p0, 0, (size_t)(2 * NGRAPH * 256 + NGRAPH) * sizeof(float), stream);
    k_count<<<(NNODES + 255) / 256, 256, 0, stream>>>(batch, cnt);

    // ---- layer 0:  h = relu(bn(relu(bn((x+agg)@W1+b1))@W2+b2))
    k_copy4<<<(NNODES * 128 / 4 + 255) / 256, 256, 0, stream>>>(
        (float4*)buf0, (const float4*)x, NNODES * 128 / 4);
    k_scatter_add<128><<<(int)(((long long)NEDGES * 128 + 255) / 256), 256, 0, stream>>>(
        x, src, dst, buf0);
    k_gemm_bn_relu<128><<<NNODES / 64, 256, 0, stream>>>(
        buf0, l0_W1, l0_b1, l0_g1, l0_be1, buf1);
    k_gemm_bn_relu<256><<<NNODES / 64, 256, 0, stream>>>(
        buf1, l0_W2, l0_b2, l0_g2, l0_be2, buf2);
    k_pool<<<(int)(((long long)NNODES * 256 + 255) / 256), 256, 0, stream>>>(
        buf2, batch, p0);

    // ---- layer 1
    k_copy4<<<(NNODES * 256 / 4 + 255) / 256, 256, 0, stream>>>(
        (float4*)buf0, (const float4*)buf2, NNODES * 256 / 4);
    k_scatter_add<256><<<(int)(((long long)NEDGES * 256 + 255) / 256), 256, 0, stream>>>(
        buf2, src, dst, buf0);
    k_gemm_bn_relu<256><<<NNODES / 64, 256, 0, stream>>>(
        buf0, l1_W1, l1_b1, l1_g1, l1_be1, buf1);
    k_gemm_bn_relu<256><<<NNODES / 64, 256, 0, stream>>>(
        buf1, l1_W2, l1_b2, l1_g2, l1_be2, buf2);
    k_pool<<<(int)(((long long)NNODES * 256 + 255) / 256), 256, 0, stream>>>(
        buf2, batch, p1);

    // ---- readout
    k_head<<<1, 64, 0, stream>>>(p0, p1, cnt, lin_W, lin_b, (float*)d_out);
}